// RingPolicyEstimator_80032420594065
// MI455X (gfx1250) — compile-verified
//
#include <hip/hip_runtime.h>
#include <hip/hip_bf16.h>

// ---------------------------------------------------------------------------
// Problem constants (from reference)
// ---------------------------------------------------------------------------
static constexpr int N_NODES = 512;
static constexpr int B_GR    = 64;
static constexpr int HID     = 16;
static constexpr int E_EDGE  = 262144;
static constexpr int N_TOTAL = B_GR * N_NODES;        // 32768
static constexpr int OUT_ROW = N_NODES * N_NODES + 1; // 262145 (odd!)

typedef __attribute__((ext_vector_type(2))) float v2f;
typedef __attribute__((ext_vector_type(8))) float v8f;

// ---------------------------------------------------------------------------
// 1) degree = 1 (self loop)  ... then += 1 per incoming edge ... then rsqrt
// ---------------------------------------------------------------------------
__global__ void deg_init_kernel(float* __restrict__ deg) {
    int i = blockIdx.x * blockDim.x + threadIdx.x;
    if (i < N_TOTAL) deg[i] = 1.0f;
}

__global__ void deg_accum_kernel(const int* __restrict__ ei, float* __restrict__ deg) {
    int e = blockIdx.x * blockDim.x + threadIdx.x;
    if (e < E_EDGE) {
        int d = ei[2 * e + 1];              // searchsorted(arange, v) == v
        atomicAdd(&deg[d], 1.0f);
    }
}

__global__ void dinv_kernel(const float* __restrict__ deg, float* __restrict__ dinv) {
    int i = blockIdx.x * blockDim.x + threadIdx.x;
    if (i < N_TOTAL) dinv[i] = rsqrtf(deg[i]);   // deg >= 1 always (self loop)
}

// ---------------------------------------------------------------------------
// 2) Fused tall-skinny GEMM: h_act = x @ W_act ; h_edge = x @ W_edge
//    x: 32768x512 f32, W: 512x16 f32.  One wave -> one 16-row tile of x,
//    accumulating both 16x16 f32 outputs with V_WMMA_F32_16X16X4_F32.
//
//    ISA 32-bit A 16x4 layout: lane = M (mod 16); VGPR0 holds K = kh,
//    VGPR1 holds K = kh+1, where kh = (lane>=16 ? 2 : 0) within the 4-step.
//    B 4x16 mirrors: lane = N (mod 16), VGPR pair = {K=kh, K=kh+1}.
//    C/D 16x16: VGPR v, lane l -> M = v + (l>=16 ? 8:0), N = l%16.
// ---------------------------------------------------------------------------
__global__ void gemm_h_kernel(const float* __restrict__ x,
                              const float* __restrict__ Wa,
                              const float* __restrict__ We,
                              float* __restrict__ ha,
                              float* __restrict__ he) {
    const int lane  = threadIdx.x & 31;
    const int wave  = threadIdx.x >> 5;
    const int mtile = blockIdx.x * 4 + wave;          // 0 .. 2047
    const int mbase = mtile * 16;

    const int l15   = lane & 15;
    const int khalf = (lane >> 4) << 1;               // 0 or 2

    const float* __restrict__ xrow = x + (size_t)(mbase + l15) * 512;

    v8f ca = {};
    v8f ce = {};
    for (int k = 0; k < 512; k += 4) {
        const int kk = k + khalf;
        v2f a = *(const v2f*)(xrow + kk);             // x[M, kk], x[M, kk+1]
        v2f ba, be;
        ba.x = Wa[(size_t)kk * HID + l15];
        ba.y = Wa[(size_t)(kk + 1) * HID + l15];
        be.x = We[(size_t)kk * HID + l15];
        be.y = We[(size_t)(kk + 1) * HID + l15];
        ca = __builtin_amdgcn_wmma_f32_16x16x4_f32(false, a, false, ba,
                                                   (short)0, ca, false, false);
        ce = __builtin_amdgcn_wmma_f32_16x16x4_f32(false, a, false, be,
                                                   (short)0, ce, false, false);
    }

    const int moff = (lane >> 4) << 3;                // 0 or 8
#pragma unroll
    for (int v = 0; v < 8; ++v) {
        const int m = mbase + moff + v;
        ha[(size_t)m * HID + l15] = ca[v];
        he[(size_t)m * HID + l15] = ce[v];
    }
}

// ---------------------------------------------------------------------------
// 3a) self-loop + bias:  g = dinv[n]^2 * h[n] + b
// ---------------------------------------------------------------------------
__global__ void init_out_kernel(const float* __restrict__ ha,
                                const float* __restrict__ he,
                                const float* __restrict__ dinv,
                                const float* __restrict__ ba,
                                const float* __restrict__ be,
                                float* __restrict__ ga,
                                float* __restrict__ ge) {
    int t = blockIdx.x * blockDim.x + threadIdx.x;    // N_TOTAL*16 threads
    if (t >= N_TOTAL * HID) return;
    int n = t >> 4;
    int f = t & 15;
    float w = dinv[n] * dinv[n];
    ga[t] = w * ha[t] + ba[f];
    ge[t] = w * he[t] + be[f];
}

// ---------------------------------------------------------------------------
// 3b) edge scatter: g[dst] += dinv[src]*dinv[dst] * h[src]
//     one thread per (edge, feature); handles both GCNs.
// ---------------------------------------------------------------------------
__global__ void scatter_kernel(const int* __restrict__ ei,
                               const float* __restrict__ dinv,
                               const float* __restrict__ ha,
                               const float* __restrict__ he,
                               float* __restrict__ ga,
                               float* __restrict__ ge) {
    long long t = (long long)blockIdx.x * blockDim.x + threadIdx.x;
    if (t >= (long long)E_EDGE * HID) return;
    int e = (int)(t >> 4);
    int f = (int)(t & 15);
    int s = ei[2 * e];
    int d = ei[2 * e + 1];
    float nrm = dinv[s] * dinv[d];
    atomicAdd(&ga[(size_t)d * HID + f], nrm * ha[(size_t)s * HID + f]);
    atomicAdd(&ge[(size_t)d * HID + f], nrm * he[(size_t)s * HID + f]);
}

// ---------------------------------------------------------------------------
// 4) at[b] = sum over 512 nodes of mean over 16 feats = (1/16)*sum(all)
//    One 256-thread block per graph; out stride is OUT_ROW.
// ---------------------------------------------------------------------------
__global__ void at_reduce_kernel(const float* __restrict__ ga,
                                 float* __restrict__ out) {
    __shared__ float red[256];
    const int b = blockIdx.x;
    const float* __restrict__ g = ga + (size_t)b * N_NODES * HID;
    float s = 0.0f;
    for (int i = threadIdx.x; i < N_NODES * HID; i += 256) s += g[i];
    red[threadIdx.x] = s;
    __syncthreads();
    for (int w = 128; w > 0; w >>= 1) {
        if (threadIdx.x < w) red[threadIdx.x] += red[threadIdx.x + w];
        __syncthreads();
    }
    if (threadIdx.x == 0) out[(size_t)b * OUT_ROW] = red[0] * (1.0f / 16.0f);
}

// ---------------------------------------------------------------------------
// 5) Gram: pair[b][n][m] = sum_f eh[b][n][f] * eh[b][m][f]
//    D = A x B with A[M=n][K=f], B[K=f][N=m] = eh[m][f].
//    One wave per 16x16 output tile, 4x V_WMMA_F32_16X16X4_F32 (K=16).
//    Scalar f32 stores (row base b*262145+1 has no 16B alignment guarantee).
// ---------------------------------------------------------------------------
__global__ void gram_kernel(const float* __restrict__ eh,
                            float* __restrict__ out) {
    const int lane = threadIdx.x & 31;
    const int wave = threadIdx.x >> 5;
    const int tile = blockIdx.x * 4 + wave;           // 0 .. 65535
    const int b     = tile >> 10;                     // graph
    const int t     = tile & 1023;
    const int ntile = t >> 5;                         // output-row tile
    const int mtile = t & 31;                         // output-col tile

    const float* __restrict__ ehb = eh + (size_t)b * N_NODES * HID;
    const int l15   = lane & 15;
    const int khalf = (lane >> 4) << 1;               // 0 or 2

    v8f c = {};
#pragma unroll
    for (int k = 0; k < HID; k += 4) {
        const int kk = k + khalf;
        v2f a  = *(const v2f*)(ehb + (size_t)(ntile * 16 + l15) * HID + kk);
        v2f bb = *(const v2f*)(ehb + (size_t)(mtile * 16 + l15) * HID + kk);
        c = __builtin_amdgcn_wmma_f32_16x16x4_f32(false, a, false, bb,
                                                  (short)0, c, false, false);
    }

    float* __restrict__ ob = out + (size_t)b * OUT_ROW + 1;
    const int moff = (lane >> 4) << 3;                // 0 or 8
#pragma unroll
    for (int v = 0; v < 8; ++v) {
        const int n = ntile * 16 + moff + v;          // output row
        const int m = mtile * 16 + l15;               // output col
        ob[(size_t)n * N_NODES + m] = c[v];
    }
}

// ---------------------------------------------------------------------------
// Host launcher
// ---------------------------------------------------------------------------
extern "C" void kernel_launch(void* const* d_in, const int* in_sizes, int n_in,
                              void* d_out, int out_size, void* d_ws, size_t ws_size,
                              hipStream_t stream) {
    const float* x   = (const float*)d_in[0];   // node_feature (32768, 512)
    const float* Wa  = (const float*)d_in[1];   // W_act  (512, 16)
    const float* ba  = (const float*)d_in[2];   // b_act  (16,)
    const float* We  = (const float*)d_in[3];   // W_edge (512, 16)
    const float* be  = (const float*)d_in[4];   // b_edge (16,)
    const int*   ei  = (const int*)d_in[5];     // edge_index (E, 2)
    // d_in[6] node_index (identity), d_in[7] batch_ptr (uniform) -- not needed
    float* out = (float*)d_out;

    // workspace layout (floats)
    float* ws   = (float*)d_ws;
    float* deg  = ws;                            // N_TOTAL
    float* dinv = deg  + N_TOTAL;                // N_TOTAL
    float* ha   = dinv + N_TOTAL;                // N_TOTAL*16
    float* he   = ha   + (size_t)N_TOTAL * HID;  // N_TOTAL*16
    float* ga   = he   + (size_t)N_TOTAL * HID;  // N_TOTAL*16
    float* ge   = ga   + (size_t)N_TOTAL * HID;  // N_TOTAL*16

    // 1) degrees & normalization
    deg_init_kernel<<<N_TOTAL / 256, 256, 0, stream>>>(deg);
    deg_accum_kernel<<<E_EDGE / 256, 256, 0, stream>>>(ei, deg);
    dinv_kernel<<<N_TOTAL / 256, 256, 0, stream>>>(deg, dinv);

    // 2) h = x @ {W_act, W_edge}  (WMMA f32, one wave per 16-row tile)
    gemm_h_kernel<<<(N_TOTAL / 16) / 4, 128, 0, stream>>>(x, Wa, We, ha, he);

    // 3) GCN aggregate: self loop + bias, then edge scatter
    init_out_kernel<<<(N_TOTAL * HID) / 256, 256, 0, stream>>>(ha, he, dinv,
                                                               ba, be, ga, ge);
    scatter_kernel<<<(E_EDGE * HID) / 256, 256, 0, stream>>>(ei, dinv, ha, he,
                                                             ga, ge);

    // 4) at -> out[:, 0]
    at_reduce_kernel<<<B_GR, 256, 0, stream>>>(ga, out);

    // 5) Gram matrices -> out[:, 1:]
    gram_kernel<<<(B_GR * 32 * 32) / 4, 128, 0, stream>>>(ge, out);
}